// ColourLoss_36172214567798
// MI455X (gfx1250) — compile-verified
//
#include <hip/hip_runtime.h>

typedef __attribute__((ext_vector_type(2))) float v2f;
typedef __attribute__((ext_vector_type(8))) float v8f;

#define KBINS   255
#define NPIX    (256 * 256)          // pixels per channel image
#define PARTS   16                   // workgroups per channel slice
#define PIX_PER_WG (NPIX / PARTS)    // 4096
#define RADIUS  9                    // sigmoid window support in bins

__device__ __forceinline__ float sigm(float z) {
    // 1/(1+e^-z): v_exp_f32 + v_rcp_f32
    return __builtin_amdgcn_rcpf(1.0f + __expf(-z));
}

// ---------------------------------------------------------------------------
// Kernel 0: zero the 32x256 histogram workspace (harness poisons d_ws)
// ---------------------------------------------------------------------------
__global__ void zero_kernel(float* __restrict__ p, int n) {
    int i = blockIdx.x * blockDim.x + threadIdx.x;
    if (i < n) p[i] = 0.0f;
}

// ---------------------------------------------------------------------------
// Kernel 1: soft histograms.
// slice = imgsel*16 + batch*2 + ch01  (ch01: 0 = green(ch1), 1 = blue(ch2))
// Red channel skipped: emd(r_hist, r_hist) == 0 exactly.
// Telescoped edges: pj_k = sigm(2.5*(t-k)+1.25) - sigm(2.5*(t-(k+1))+1.25),
// with t = 255*x - 0.5; adjacent bins share one sigmoid evaluation.
// ---------------------------------------------------------------------------
__global__ void hist_kernel(const float* __restrict__ img,
                            const float* __restrict__ img_t,
                            float* __restrict__ hist /* [32][256] */) {
    __shared__ float sh[256];
    const int tid    = threadIdx.x;
    const int slice  = blockIdx.x / PARTS;
    const int part   = blockIdx.x % PARTS;
    const int imgsel = slice >> 4;
    const int v      = slice & 15;        // batch*2 + ch01
    const int b      = v >> 1;
    const int c      = (v & 1) + 1;       // channel index 1 (g) or 2 (b)

    const float* src = (imgsel ? img_t : img)
                     + ((size_t)(b * 3 + c) << 16)   // * 65536
                     + (size_t)part * PIX_PER_WG;

    sh[tid] = 0.0f;
    __syncthreads();

    #pragma unroll 1
    for (int i = 0; i < PIX_PER_WG / 256; ++i) {
        if (i + 1 < PIX_PER_WG / 256)
            __builtin_prefetch(src + (i + 1) * 256 + tid, 0, 3);
        float x = src[i * 256 + tid];
        float t = __builtin_fmaf(x, 255.0f, -0.5f);
        int j0  = (int)floorf(t + 0.5f);             // nearest bin, 0..254
        int klo = max(j0 - RADIUS, 0);
        int khi = min(j0 + RADIUS, KBINS - 1);
        float eprev = sigm(__builtin_fmaf(2.5f, t - (float)klo, 1.25f));
        for (int k = klo; k <= khi; ++k) {
            float enext = sigm(__builtin_fmaf(2.5f, t - (float)(k + 1), 1.25f));
            atomicAdd(&sh[k], eprev - enext);        // ds_add_f32
            eprev = enext;
        }
    }
    __syncthreads();
    atomicAdd(&hist[slice * 256 + tid], sh[tid]);    // global_atomic_add_f32
}

// ---------------------------------------------------------------------------
// Kernel 2: EMD via WMMA prefix-sum.
// CDF(16x256) = DIFF(16x256) @ T(256x256), T[j,k] = (j <= k).
// One wave32; V_WMMA_F32_16X16X4_F32, A = 16x4 diff slab, B = 4x16
// triangular-ones fragment generated in registers. Squares of the f32
// accumulator tiles (masking padded column k=255) are reduced over N lanes
// to produce loss[b] = emd_g(b) + emd_b(b).
// ---------------------------------------------------------------------------
__global__ void emd_wmma_kernel(const float* __restrict__ hist,
                                float* __restrict__ out /* [8] */) {
    __shared__ float D[16][257];      // pad to kill bank conflicts
    __shared__ float red[8][32];
    const int lane = threadIdx.x;     // wave32, block == 1 wave, EXEC all-1s
    const float invN = 1.0f / (float)NPIX;

    // diff rows: v = batch*2 + ch01 ; slice v = img, slice 16+v = img_t
    for (int idx = lane; idx < 16 * 256; idx += 32) {
        int v = idx >> 8, k = idx & 255;
        D[v][k] = (hist[v * 256 + k] - hist[(16 + v) * 256 + k]) * invN;
    }
    __syncthreads();

    const int mrow = lane & 15;                // M for A, N for B/C/D
    const int koff = (lane < 16) ? 0 : 2;      // K pair held by this lane

    float acc[8];
    #pragma unroll
    for (int i = 0; i < 8; ++i) acc[i] = 0.0f;

    for (int kt = 0; kt < 16; ++kt) {          // output column tile
        v8f cfrag = {};
        const int ncol = kt * 16 + mrow;       // this lane's output column
        for (int j4 = 0; j4 < (kt + 1) * 4; ++j4) {   // K chunks of 4
            const int jb = j4 * 4 + koff;
            v2f a, bm;
            a.x  = D[mrow][jb];
            a.y  = D[mrow][jb + 1];
            bm.x = (jb     <= ncol) ? 1.0f : 0.0f;    // T[j, ncol]
            bm.y = (jb + 1 <= ncol) ? 1.0f : 0.0f;
            cfrag = __builtin_amdgcn_wmma_f32_16x16x4_f32(
                false, a, false, bm, (short)0, cfrag, false, false);
        }
        // mask out padded column k == 255 (cumsum runs over 255 bins only)
        float msk = (ncol < KBINS) ? 1.0f : 0.0f;
        #pragma unroll
        for (int i = 0; i < 8; ++i) acc[i] += cfrag[i] * cfrag[i] * msk;
    }

    #pragma unroll
    for (int i = 0; i < 8; ++i) red[i][lane] = acc[i];
    __syncthreads();

    if (lane < 8) {
        const int bb = lane;
        float s = 0.0f;
        for (int m = 2 * bb; m <= 2 * bb + 1; ++m) {   // g row, b row
            int i  = (m < 8) ? m : (m - 8);            // C/D VGPR row
            int l0 = (m < 8) ? 0 : 16;                 // lane half for this M
            for (int l = 0; l < 16; ++l) s += red[i][l0 + l];
        }
        out[bb] = s;   // + emd(r,r) which is exactly 0
    }
}

// ---------------------------------------------------------------------------
extern "C" void kernel_launch(void* const* d_in, const int* in_sizes, int n_in,
                              void* d_out, int out_size, void* d_ws, size_t ws_size,
                              hipStream_t stream) {
    const float* img   = (const float*)d_in[0];
    const float* img_t = (const float*)d_in[1];
    float* hist = (float*)d_ws;                 // 32 * 256 * 4B = 32 KB
    float* out  = (float*)d_out;                // 8 floats

    zero_kernel<<<32, 256, 0, stream>>>(hist, 32 * 256);
    hist_kernel<<<32 * PARTS, 256, 0, stream>>>(img, img_t, hist);
    emd_wmma_kernel<<<1, 32, 0, stream>>>(hist, out);
}